// PureAddShiftMP_88785563942923
// MI455X (gfx1250) — compile-verified
//
#include <hip/hip_runtime.h>
#include <stdint.h>

// out_g[n,o,y,x] = sum_{k=0..2} x[n, 3o+k, y+dy, x+dx], (dy,dx)=pos[(k+g)%3]
// pos = {(-1,-1), (-1,0), (0,-1)}, zero outside the image.
// Pure shift-gather-add: memory-bound (~201 MB traffic -> ~8.6us @ 23.3 TB/s).
// Strategy: stage each input channel into LDS exactly once via CDNA5 async
// global->LDS B128 transfers (ASYNCcnt path), resolve the 3x shifted reuse in
// LDS, write the three group outputs with nontemporal B128 stores.

typedef float v4f __attribute__((ext_vector_type(4)));

#define CIN        768
#define COUT       256
#define HW         64
#define TROWS      65      // row 0 = zero pad (dy=-1 at y=0)
#define TSTRIDE    68      // 4-float left pad keeps data 16B-aligned; col 3 = zero pad
#define NBLK_PER_N COUT

__device__ __forceinline__ uint32_t lds_addr32(const void* p) {
  // flat shared pointer = {aperture_hi, lds_offset}; DS/async ops want the low 32 bits
  return (uint32_t)(uintptr_t)p;
}

__device__ __forceinline__ void async_g2l_b128(uint32_t lds_off, const float* gptr) {
  asm volatile("global_load_async_to_lds_b128 %0, %1, off"
               :: "v"(lds_off), "v"(gptr)
               : "memory");
}

__device__ __forceinline__ void wait_async_zero() {
  asm volatile("s_wait_asynccnt 0" ::: "memory");
}

__global__ __launch_bounds__(256)
void PureAddShiftMP_shiftadd_kernel(const float* __restrict__ x,
                                    float* __restrict__ out) {
  __shared__ float tile[3][TROWS][TSTRIDE];

  const int t   = threadIdx.x;
  const int blk = blockIdx.x;        // 0 .. 8*256-1
  const int n   = blk >> 8;
  const int oc  = blk & (COUT - 1);

  // ---- zero the padding: top row (r=0, all cols) and left col (c=3, r=1..64)
  for (int i = t; i < 3 * TSTRIDE; i += 256)
    tile[i / TSTRIDE][0][i % TSTRIDE] = 0.0f;
  for (int i = t; i < 3 * HW; i += 256)
    tile[i >> 6][(i & 63) + 1][3] = 0.0f;

  // ---- async global -> LDS: 3 channels x 64 rows x 16 float4 = 3072 B128 ops
  const float* gbase = x + (size_t)(n * CIN + 3 * oc) * (HW * HW);
#pragma unroll
  for (int i = 0; i < 12; ++i) {
    const int f   = t + i * 256;        // 0..3071
    const int ch  = f >> 10;            // /1024
    const int rem = f & 1023;
    const int y   = rem >> 4;
    const int j   = rem & 15;
    const float* g = gbase + ch * (HW * HW) + y * HW + 4 * j;
    async_g2l_b128(lds_addr32(&tile[ch][y + 1][4 + 4 * j]), g);
  }
  wait_async_zero();      // this wave's async transfers landed in LDS
  __syncthreads();        // everyone's transfers + pad stores visible

  // ---- compute: 3 output planes of 64x64, 4 float4 per thread per plane
  const size_t obase = (size_t)(n * COUT + oc) * (HW * HW);
  const size_t gsz   = (size_t)8 * COUT * HW * HW;   // elements per output group

#pragma unroll
  for (int i = 0; i < 4; ++i) {
    const int f  = t + i * 256;         // 0..1023
    const int y  = f >> 4;
    const int x0 = (f & 15) * 4;
    v4f o0, o1, o2;
#pragma unroll
    for (int e = 0; e < 4; ++e) {
      const int c = x0 + e;
      // A=(-1,-1) -> tile[y][c+3]; B=(-1,0) -> tile[y][c+4]; C=(0,-1) -> tile[y+1][c+3]
      const float A0 = tile[0][y][c + 3], B0 = tile[0][y][c + 4], C0 = tile[0][y + 1][c + 3];
      const float A1 = tile[1][y][c + 3], B1 = tile[1][y][c + 4], C1 = tile[1][y + 1][c + 3];
      const float A2 = tile[2][y][c + 3], B2 = tile[2][y][c + 4], C2 = tile[2][y + 1][c + 3];
      o0[e] = A0 + B1 + C2;   // group 0: k -> pos[k]
      o1[e] = B0 + C1 + A2;   // group 1: k -> pos[(k+1)%3]
      o2[e] = C0 + A1 + B2;   // group 2: k -> pos[(k+2)%3]
    }
    const size_t off = obase + (size_t)y * HW + x0;
    __builtin_nontemporal_store(o0, (v4f*)(out + off));
    __builtin_nontemporal_store(o1, (v4f*)(out + gsz + off));
    __builtin_nontemporal_store(o2, (v4f*)(out + 2 * gsz + off));
  }
}

extern "C" void kernel_launch(void* const* d_in, const int* in_sizes, int n_in,
                              void* d_out, int out_size, void* d_ws, size_t ws_size,
                              hipStream_t stream) {
  (void)in_sizes; (void)n_in; (void)d_ws; (void)ws_size; (void)out_size;
  const float* x = (const float*)d_in[0];   // (8, 768, 64, 64) f32
  float* out = (float*)d_out;               // 3 x (8, 256, 64, 64) f32, concat
  PureAddShiftMP_shiftadd_kernel<<<dim3(8 * NBLK_PER_N), dim3(256), 0, stream>>>(x, out);
}